// ProtDiGCNEncoderDecoder_minibatch_11570641895933
// MI455X (gfx1250) — compile-verified
//
#include <hip/hip_runtime.h>
#include <hip/hip_bf16.h>
#include <math.h>

typedef float v2f __attribute__((ext_vector_type(2)));
typedef float v8f __attribute__((ext_vector_type(8)));

#define CHANNELS 128

// ---------------------------------------------------------------------------
// Degree / normalization
// ---------------------------------------------------------------------------
__global__ __launch_bounds__(256) void deg_init_k(float* __restrict__ deg, int n) {
    int i = blockIdx.x * 256 + threadIdx.x;
    if (i < n) deg[i] = 1.0f;  // self-loop contributes 1
}

__global__ __launch_bounds__(256) void deg_accum_k(const int* __restrict__ ei,
                                                   float* __restrict__ deg,
                                                   int nedges) {
    int e = blockIdx.x * 256 + threadIdx.x;
    if (e < nedges) {
        int c = ei[(size_t)nedges + e];  // col = edge_index[1][e]
        atomicAdd(&deg[c], 1.0f);
    }
}

__global__ __launch_bounds__(256) void dinv_k(const float* __restrict__ deg,
                                              float* __restrict__ dinv, int n) {
    int i = blockIdx.x * 256 + threadIdx.x;
    if (i < n) dinv[i] = rsqrtf(deg[i]);  // deg >= 1 always
}

// ---------------------------------------------------------------------------
// GEMM: Y[n, NCOL] = X[n, 128] @ W[128, NCOL]   (fp32 WMMA 16x16x4)
// Block = 256 threads = 8 waves; wave w owns rows [blk*128 + 16w, +16).
// W is staged in LDS as interleaved K-pairs so each B fragment is a single
// aligned 8-byte ds_load_b64 (no repacking movs):
//   sW2[kp][c] = ( W[k0+2kp][c], W[k0+2kp+1][c] )
// Row pad of 16 float2 => row stride 288 floats == 32 mod 64 banks, so the
// two half-waves (which read adjacent pair-rows) land on disjoint bank halves.
// ---------------------------------------------------------------------------
template <int NCOL>
__global__ __launch_bounds__(256) void gemm_wmma_k(const float* __restrict__ X,
                                                   const float* __restrict__ W,
                                                   float* __restrict__ Y, int nrows) {
    constexpr int NT   = NCOL / 16;    // output col tiles per wave
    constexpr int PAD2 = 16;           // pad in float2 units
    constexpr int LDW2 = NCOL + PAD2;  // row stride in float2 units
    __shared__ v2f sW2[16 * LDW2];

    const int tid  = threadIdx.x;
    const int lane = tid & 31;
    const int lm   = lane & 15;        // l % 16
    const int lh   = lane >> 4;        // l / 16
    const int wave = tid >> 5;
    const int rowBase = blockIdx.x * 128 + wave * 16;
    int r = rowBase + lm;
    const int rc = (r < nrows) ? r : (nrows - 1);   // clamp loads, guard stores

    v8f acc[NT] = {};

    for (int k0 = 0; k0 < CHANNELS; k0 += 32) {
        __syncthreads();
        // cooperative stage of W chunk [k0..k0+32) x NCOL as K-pair float2s
        for (int i = tid; i < 16 * NCOL; i += 256) {
            int kp = i / NCOL;             // pair row 0..15
            int cc = i - kp * NCOL;
            v2f w2;
            w2.x = W[(size_t)(k0 + 2 * kp) * NCOL + cc];
            w2.y = W[(size_t)(k0 + 2 * kp + 1) * NCOL + cc];
            sW2[kp * LDW2 + cc] = w2;
        }
        __syncthreads();

        // A fragments for this K-chunk: a[ks].{x,y} = X[row][k0+4ks+2lh+{0,1}]
        v2f a[8];
#pragma unroll
        for (int ks = 0; ks < 8; ++ks) {
            const float* p = X + (size_t)rc * CHANNELS + k0 + 4 * ks + 2 * lh;
            a[ks] = *(const v2f*)p;   // 8B global load
        }

#pragma unroll
        for (int t = 0; t < NT; ++t) {
#pragma unroll
            for (int ks = 0; ks < 8; ++ks) {
                // kr = 4ks + 2lh (even) -> pair row = 2ks + lh
                v2f b = sW2[(2 * ks + lh) * LDW2 + t * 16 + lm];
                acc[t] = __builtin_amdgcn_wmma_f32_16x16x4_f32(
                    false, a[ks], false, b, (short)0, acc[t], false, false);
            }
        }
    }

    // D layout: D[v + 8*lh][lm]
#pragma unroll
    for (int t = 0; t < NT; ++t) {
#pragma unroll
        for (int v = 0; v < 8; ++v) {
            int rr = rowBase + v + 8 * lh;
            if (rr < nrows) Y[(size_t)rr * NCOL + t * 16 + lm] = acc[t][v];
        }
    }
}

// ---------------------------------------------------------------------------
// Self-loop term: Acc[i][c] = H[i][c] * dinv[i]^2
// ---------------------------------------------------------------------------
__global__ __launch_bounds__(256) void selfloop_init_k(const float* __restrict__ H,
                                                       const float* __restrict__ dinv,
                                                       float* __restrict__ Acc, int n) {
    int i = blockIdx.x * 256 + threadIdx.x;
    if (i < n * CHANNELS) {
        int row = i >> 7;
        float d = dinv[row];
        Acc[i] = H[i] * d * d;
    }
}

// ---------------------------------------------------------------------------
// Edge scatter: one wave per edge, float4 per lane, atomic f32 adds.
// Acc[col] += H[row] * dinv[row]*dinv[col]
// ---------------------------------------------------------------------------
__global__ __launch_bounds__(256) void scatter_edges_k(const int* __restrict__ ei,
                                                       const float* __restrict__ dinv,
                                                       const float* __restrict__ H,
                                                       float* __restrict__ Acc,
                                                       int nedges) {
    int e    = blockIdx.x * 8 + (threadIdx.x >> 5);
    int lane = threadIdx.x & 31;
    if (e >= nedges) return;
    int r = ei[e];
    int c = ei[(size_t)nedges + e];
    float w = dinv[r] * dinv[c];
    const float4 v = ((const float4*)(H + (size_t)r * CHANNELS))[lane];
    float* p = Acc + (size_t)c * CHANNELS + lane * 4;
    atomicAdd(p + 0, v.x * w);
    atomicAdd(p + 1, v.y * w);
    atomicAdd(p + 2, v.z * w);
    atomicAdd(p + 3, v.w * w);
}

// ---------------------------------------------------------------------------
// Bias (+ optional ReLU)
// ---------------------------------------------------------------------------
template <int RELU>
__global__ __launch_bounds__(256) void bias_act_k(float* __restrict__ Acc,
                                                  const float* __restrict__ b, int n) {
    int i = blockIdx.x * 256 + threadIdx.x;
    if (i < n * CHANNELS) {
        float v = Acc[i] + b[i & (CHANNELS - 1)];
        Acc[i] = RELU ? fmaxf(v, 0.0f) : v;
    }
}

// ---------------------------------------------------------------------------
// Row-wise L2 normalize (wave per row, 128 ch)
// ---------------------------------------------------------------------------
__global__ __launch_bounds__(256) void l2norm_k(const float* __restrict__ H,
                                                float* __restrict__ Emb, int n) {
    int row  = blockIdx.x * 8 + (threadIdx.x >> 5);
    int lane = threadIdx.x & 31;
    if (row >= n) return;
    float4 v = ((const float4*)(H + (size_t)row * CHANNELS))[lane];
    float s = v.x * v.x + v.y * v.y + v.z * v.z + v.w * v.w;
#pragma unroll
    for (int off = 16; off > 0; off >>= 1) s += __shfl_xor(s, off, 32);
    float inv = 1.0f / (sqrtf(s) + 1e-12f);
    float4 o = make_float4(v.x * inv, v.y * inv, v.z * inv, v.w * inv);
    ((float4*)(Emb + (size_t)row * CHANNELS))[lane] = o;
}

// ---------------------------------------------------------------------------
// In-place bias + log_softmax over 64 logits (wave per row, 2 vals/lane)
// ---------------------------------------------------------------------------
__global__ __launch_bounds__(256) void logsoftmax_k(float* __restrict__ L,
                                                    const float* __restrict__ bd, int n) {
    int row  = blockIdx.x * 8 + (threadIdx.x >> 5);
    int lane = threadIdx.x & 31;
    if (row >= n) return;
    float2 x = ((const float2*)(L + (size_t)row * 64))[lane];
    x.x += bd[lane * 2];
    x.y += bd[lane * 2 + 1];
    float m = fmaxf(x.x, x.y);
#pragma unroll
    for (int off = 16; off > 0; off >>= 1) m = fmaxf(m, __shfl_xor(m, off, 32));
    float s = __expf(x.x - m) + __expf(x.y - m);
#pragma unroll
    for (int off = 16; off > 0; off >>= 1) s += __shfl_xor(s, off, 32);
    float ls = m + logf(s);
    float2 o = make_float2(x.x - ls, x.y - ls);
    ((float2*)(L + (size_t)row * 64))[lane] = o;
}

// ---------------------------------------------------------------------------
// Launch
// ---------------------------------------------------------------------------
extern "C" void kernel_launch(void* const* d_in, const int* in_sizes, int n_in,
                              void* d_out, int out_size, void* d_ws, size_t ws_size,
                              hipStream_t stream) {
    const float* x  = (const float*)d_in[0];
    const int*   ei = (const int*)d_in[1];  // [2, E]
    const float* W1 = (const float*)d_in[2];
    const float* b1 = (const float*)d_in[3];
    const float* W2 = (const float*)d_in[4];
    const float* b2 = (const float*)d_in[5];
    const float* Wd = (const float*)d_in[6];
    const float* bd = (const float*)d_in[7];

    const int N = in_sizes[0] / CHANNELS;
    const int E = in_sizes[1] / 2;

    float* ws   = (float*)d_ws;
    float* deg  = ws;
    float* dinv = ws + N;
    float* bufA = ws + 2 * (size_t)N;                 // N x 128
    float* bufB = bufA + (size_t)N * CHANNELS;        // N x 128

    float* logits = (float*)d_out;                    // N x 64   (output 0)
    float* emb    = (float*)d_out + (size_t)N * 64;   // N x 128  (output 1)

    const int nElem = N * CHANNELS;
    dim3 blk(256);
    dim3 gN((N + 255) / 256);
    dim3 gE((E + 255) / 256);
    dim3 gElem((nElem + 255) / 256);
    dim3 gGemm((N + 127) / 128);
    dim3 gRow((N + 7) / 8);
    dim3 gEdge((E + 7) / 8);

    // normalization coefficients
    deg_init_k<<<gN, blk, 0, stream>>>(deg, N);
    deg_accum_k<<<gE, blk, 0, stream>>>(ei, deg, E);
    dinv_k<<<gN, blk, 0, stream>>>(deg, dinv, N);

    // ---- layer 1: h = relu(aggregate(x @ W1) + b1)
    gemm_wmma_k<128><<<gGemm, blk, 0, stream>>>(x, W1, bufA, N);
    selfloop_init_k<<<gElem, blk, 0, stream>>>(bufA, dinv, bufB, N);
    scatter_edges_k<<<gEdge, blk, 0, stream>>>(ei, dinv, bufA, bufB, E);
    bias_act_k<1><<<gElem, blk, 0, stream>>>(bufB, b1, N);

    // ---- layer 2: h = aggregate(h @ W2) + b2
    gemm_wmma_k<128><<<gGemm, blk, 0, stream>>>(bufB, W2, bufA, N);
    selfloop_init_k<<<gElem, blk, 0, stream>>>(bufA, dinv, bufB, N);
    scatter_edges_k<<<gEdge, blk, 0, stream>>>(ei, dinv, bufA, bufB, E);
    bias_act_k<0><<<gElem, blk, 0, stream>>>(bufB, b2, N);

    // ---- L2 normalize -> embeddings (output 1)
    l2norm_k<<<gRow, blk, 0, stream>>>(bufB, emb, N);

    // ---- decoder: logits = log_softmax(emb @ Wd + bd) (output 0)
    gemm_wmma_k<64><<<gGemm, blk, 0, stream>>>(emb, Wd, logits, N);
    logsoftmax_k<<<gRow, blk, 0, stream>>>(logits, bd, N);
}